// SchNetInteraction_4002909520406
// MI455X (gfx1250) — compile-verified
//
#include <hip/hip_runtime.h>

typedef __attribute__((ext_vector_type(2))) float v2f;
typedef __attribute__((ext_vector_type(8))) float v8f;

#define N_NODES  100000
#define N_EDGES  1600000
#define N_FEAT   64
#define EXP_FEAT 50
#define CUTOFF   5.0f
#define LOG2_F   0.69314718055994530942f
#define LDSS     68   // padded LDS row stride (floats): 4*row mod 64 spreads banks

__device__ __forceinline__ float ssp(float x) {
    // shifted softplus, numerically stable: max(x,0)+log1p(exp(-|x|)) - log2
    return fmaxf(x, 0.0f) + __logf(1.0f + __expf(-fabsf(x))) - LOG2_F;
}

__device__ __forceinline__ v8f wmma_f32(v2f a, v2f b, v8f c) {
    // V_WMMA_F32_16X16X4_F32: full fp32 precision matrix op
    return __builtin_amdgcn_wmma_f32_16x16x4_f32(
        /*neg_a=*/false, a, /*neg_b=*/false, b,
        /*c_mod=*/(short)0, c, /*reuse_a=*/false, /*reuse_b=*/false);
}

// ---------------------------------------------------------------- zero agg
__global__ void zero_kernel(float* __restrict__ p, int n) {
    int i = blockIdx.x * blockDim.x + threadIdx.x;
    if (i < n) p[i] = 0.0f;
}

// ------------------------------------------------- h = x @ lin_w.T + lin_b
__global__ __launch_bounds__(128) void hlin_kernel(
    const float* __restrict__ x, const float* __restrict__ w,
    const float* __restrict__ b, float* __restrict__ h, int n_rows)
{
    __shared__ float ws[64 * LDSS];
    __shared__ float bs[64];
    for (int i = threadIdx.x; i < 64 * 64; i += 128) {
        int n = i >> 6, k = i & 63;
        ws[n * LDSS + k] = w[i];
    }
    if (threadIdx.x < 64) bs[threadIdx.x] = b[threadIdx.x];
    __syncthreads();

    const int lane = threadIdx.x & 31, wave = threadIdx.x >> 5;
    const int hi = lane >> 4, lo = lane & 15;
    const int m0 = (blockIdx.x * 4 + wave) * 16;
    if (m0 >= n_rows) return;

    v8f c[4] = {};
    for (int kt = 0; kt < 16; ++kt) {
        const int kb = kt * 4 + hi * 2;
        v2f a = *(const v2f*)(x + (m0 + lo) * N_FEAT + kb);
#pragma unroll
        for (int nt = 0; nt < 4; ++nt) {
            v2f bb = *(const v2f*)(ws + (nt * 16 + lo) * LDSS + kb);
            c[nt] = wmma_f32(a, bb, c[nt]);
        }
    }
#pragma unroll
    for (int nt = 0; nt < 4; ++nt) {
        const int n = nt * 16 + lo;
        const float bias = bs[n];
#pragma unroll
        for (int r = 0; r < 8; ++r)
            h[(m0 + r + 8 * hi) * N_FEAT + n] = c[nt][r] + bias;
    }
}

// --- fused edge kernel: rbf -> GEMM(50,64) -> ssp -> GEMM(64,64) ->
//     msg = h[src]*filt -> atomic scatter into agg.  1 wave = 16 edges.
__global__ __launch_bounds__(128) void edge_kernel(
    const int* __restrict__ src, const int* __restrict__ dst,
    const float* __restrict__ dist,
    const float* __restrict__ fw1, const float* __restrict__ fb1,
    const float* __restrict__ fw2, const float* __restrict__ fb2,
    const float* __restrict__ h, float* __restrict__ agg)
{
    __shared__ float w1s[64 * LDSS];   // f_w1 padded to K=52 with zeros
    __shared__ float w2s[64 * LDSS];
    __shared__ float b1s[64], b2s[64];
    __shared__ float t1s[4][16 * LDSS]; // per-wave C->A relayout staging
    __shared__ int   ses[4][16], des[4][16];

    for (int i = threadIdx.x; i < 64 * 52; i += 128) {
        int n = i / 52, k = i % 52;
        w1s[n * LDSS + k] = (k < EXP_FEAT) ? fw1[n * EXP_FEAT + k] : 0.0f;
    }
    for (int i = threadIdx.x; i < 64 * 64; i += 128) {
        int n = i >> 6, k = i & 63;
        w2s[n * LDSS + k] = fw2[i];
    }
    if (threadIdx.x < 64) {
        b1s[threadIdx.x] = fb1[threadIdx.x];
        b2s[threadIdx.x] = fb2[threadIdx.x];
    }
    __syncthreads();

    const int lane = threadIdx.x & 31, wave = threadIdx.x >> 5;
    const int hi = lane >> 4, lo = lane & 15;
    const int e0 = (blockIdx.x * 4 + wave) * 16;

    if (lane < 16) {
        ses[wave][lane] = src[e0 + lane];
        des[wave][lane] = dst[e0 + lane];
    }
    const float d = dist[e0 + lo];

    const float delta = CUTOFF / (float)(EXP_FEAT - 1);
    const float coeff = -0.5f / (delta * delta);

    // GEMM1: rbf[16x52] @ w1.T  (A generated on the fly: 2 exps / lane / kt)
    v8f c[4] = {};
    for (int kt = 0; kt < 13; ++kt) {
        const int kb = kt * 4 + hi * 2;
        const float t0 = d - (float)kb * delta;
        const float t1 = d - (float)(kb + 1) * delta;
        v2f a;
        a.x = (kb     < EXP_FEAT) ? __expf(coeff * t0 * t0) : 0.0f;
        a.y = (kb + 1 < EXP_FEAT) ? __expf(coeff * t1 * t1) : 0.0f;
#pragma unroll
        for (int nt = 0; nt < 4; ++nt) {
            v2f bb = *(const v2f*)(w1s + (nt * 16 + lo) * LDSS + kb);
            c[nt] = wmma_f32(a, bb, c[nt]);
        }
    }

    // ssp(t1 + b1) -> LDS (C layout in, row-major out)
#pragma unroll
    for (int nt = 0; nt < 4; ++nt) {
        const int n = nt * 16 + lo;
        const float bias = b1s[n];
#pragma unroll
        for (int r = 0; r < 8; ++r)
            t1s[wave][(r + 8 * hi) * LDSS + n] = ssp(c[nt][r] + bias);
    }
    __builtin_amdgcn_wave_barrier();   // intra-wave LDS is in-order

    // GEMM2: act[16x64] @ w2.T
    v8f c2[4] = {};
    for (int kt = 0; kt < 16; ++kt) {
        const int kb = kt * 4 + hi * 2;
        v2f a = *(const v2f*)(&t1s[wave][lo * LDSS + kb]);
#pragma unroll
        for (int nt = 0; nt < 4; ++nt) {
            v2f bb = *(const v2f*)(w2s + (nt * 16 + lo) * LDSS + kb);
            c2[nt] = wmma_f32(a, bb, c2[nt]);
        }
    }

    // msg = h[src] * filt ; scatter-add (hits L2: h/agg are 25.6MB each)
#pragma unroll
    for (int nt = 0; nt < 4; ++nt) {
        const int n = nt * 16 + lo;
        const float bias = b2s[n];
#pragma unroll
        for (int r = 0; r < 8; ++r) {
            const int m = r + 8 * hi;
            const int s = ses[wave][m];
            const int t = des[wave][m];
            const float msg = h[s * N_FEAT + n] * (c2[nt][r] + bias);
            atomicAdd(&agg[t * N_FEAT + n], msg);
        }
    }
}

// ------------- out = ssp(agg @ m_w1.T + m_b1) @ m_w2.T + m_b2
__global__ __launch_bounds__(128) void outmlp_kernel(
    const float* __restrict__ agg,
    const float* __restrict__ w1, const float* __restrict__ b1,
    const float* __restrict__ w2, const float* __restrict__ b2,
    float* __restrict__ out, int n_rows)
{
    __shared__ float w1s[64 * LDSS], w2s[64 * LDSS];
    __shared__ float b1s[64], b2s[64];
    __shared__ float t1s[4][16 * LDSS];

    for (int i = threadIdx.x; i < 64 * 64; i += 128) {
        int n = i >> 6, k = i & 63;
        w1s[n * LDSS + k] = w1[i];
        w2s[n * LDSS + k] = w2[i];
    }
    if (threadIdx.x < 64) {
        b1s[threadIdx.x] = b1[threadIdx.x];
        b2s[threadIdx.x] = b2[threadIdx.x];
    }
    __syncthreads();

    const int lane = threadIdx.x & 31, wave = threadIdx.x >> 5;
    const int hi = lane >> 4, lo = lane & 15;
    const int m0 = (blockIdx.x * 4 + wave) * 16;
    if (m0 >= n_rows) return;

    v8f c[4] = {};
    for (int kt = 0; kt < 16; ++kt) {
        const int kb = kt * 4 + hi * 2;
        v2f a = *(const v2f*)(agg + (m0 + lo) * N_FEAT + kb);
#pragma unroll
        for (int nt = 0; nt < 4; ++nt) {
            v2f bb = *(const v2f*)(w1s + (nt * 16 + lo) * LDSS + kb);
            c[nt] = wmma_f32(a, bb, c[nt]);
        }
    }
#pragma unroll
    for (int nt = 0; nt < 4; ++nt) {
        const int n = nt * 16 + lo;
        const float bias = b1s[n];
#pragma unroll
        for (int r = 0; r < 8; ++r)
            t1s[wave][(r + 8 * hi) * LDSS + n] = ssp(c[nt][r] + bias);
    }
    __builtin_amdgcn_wave_barrier();

    v8f c2[4] = {};
    for (int kt = 0; kt < 16; ++kt) {
        const int kb = kt * 4 + hi * 2;
        v2f a = *(const v2f*)(&t1s[wave][lo * LDSS + kb]);
#pragma unroll
        for (int nt = 0; nt < 4; ++nt) {
            v2f bb = *(const v2f*)(w2s + (nt * 16 + lo) * LDSS + kb);
            c2[nt] = wmma_f32(a, bb, c2[nt]);
        }
    }
#pragma unroll
    for (int nt = 0; nt < 4; ++nt) {
        const int n = nt * 16 + lo;
        const float bias = b2s[n];
#pragma unroll
        for (int r = 0; r < 8; ++r)
            out[(m0 + r + 8 * hi) * N_FEAT + n] = c2[nt][r] + bias;
    }
}

extern "C" void kernel_launch(void* const* d_in, const int* in_sizes, int n_in,
                              void* d_out, int out_size, void* d_ws, size_t ws_size,
                              hipStream_t stream) {
    const int*   nbr   = (const int*)  d_in[0];   // [2, E]
    const float* dist  = (const float*)d_in[1];   // [E]
    const float* x     = (const float*)d_in[2];   // [N, 64]
    const float* lin_w = (const float*)d_in[3];
    const float* lin_b = (const float*)d_in[4];
    const float* fw1   = (const float*)d_in[5];
    const float* fb1   = (const float*)d_in[6];
    const float* fw2   = (const float*)d_in[7];
    const float* fb2   = (const float*)d_in[8];
    const float* mw1   = (const float*)d_in[9];
    const float* mb1   = (const float*)d_in[10];
    const float* mw2   = (const float*)d_in[11];
    const float* mb2   = (const float*)d_in[12];

    float* h   = (float*)d_ws;                       // [N,64] 25.6 MB
    float* agg = h + (size_t)N_NODES * N_FEAT;       // [N,64] 25.6 MB
    float* out = (float*)d_out;

    const int nElems = N_NODES * N_FEAT;
    zero_kernel<<<(nElems + 255) / 256, 256, 0, stream>>>(agg, nElems);

    const int nTiles = N_NODES / 16;                 // 6250
    hlin_kernel<<<(nTiles + 3) / 4, 128, 0, stream>>>(x, lin_w, lin_b, h, N_NODES);

    const int eTiles = N_EDGES / 16;                 // 100000 (exact)
    edge_kernel<<<eTiles / 4, 128, 0, stream>>>(nbr, nbr + N_EDGES, dist,
                                                fw1, fb1, fw2, fb2, h, agg);

    outmlp_kernel<<<(nTiles + 3) / 4, 128, 0, stream>>>(agg, mw1, mb1, mw2, mb2,
                                                        out, N_NODES);
}